// JTMPN_24335284699713
// MI455X (gfx1250) — compile-verified
//
#include <hip/hip_runtime.h>
#include <hip/hip_bf16.h>

// ---------------- problem constants (match reference) ----------------
#define HIDDEN    450
#define ATOM_FDIM 35
#define BOND_FDIM 5
#define BONDF     (ATOM_FDIM + BOND_FDIM)   // 40  (already %4==0, even)
#define MAX_NB    15
#define N_ATOMS   30000
#define N_BONDS   60000
#define N_MESS    10001
#define N_MOLS    1000
#define N_MSGROWS (N_MESS + N_BONDS)        // 70001
#define KH_PAD    452                       // HIDDEN padded to %4
#define KO_PAD    488                       // (ATOM_FDIM+HIDDEN)=485 padded to %4
#define NCOLPAD   512
#define NCHUNKS   32

typedef __attribute__((ext_vector_type(2))) float v2f;
typedef __attribute__((ext_vector_type(8))) float v8f;

// D(16x16,f32) = A(16x4,f32) x B(4x16,f32) + C
static __device__ __forceinline__ v8f wmma_f32(v2f a, v2f b, v8f c) {
  return __builtin_amdgcn_wmma_f32_16x16x4_f32(
      /*neg_a=*/false, a, /*neg_b=*/false, b,
      /*c_mod=*/(short)0, c, /*reuse_a=*/false, /*reuse_b=*/false);
}

// ---------------------------------------------------------------------
// fp32 WMMA GEMM:  out = act( A @ W^T  (+ addsrc) (+ bias) )
//   A  : [M x K] rows at stride lda (lda even, K%4==0, zero-padded cols)
//   W  : [N x K] row-major, stride K (zero-padded cols)
// One wave -> 32x64 tile (2 M-subtiles x 4 N-subtiles, 8 accumulators).
// 8 waves/WG -> 256x64 block tile. Software-pipelined inner loop:
// prefetch next k-step's 6 float2 fragments, then issue 8 WMMAs.
// ---------------------------------------------------------------------
__global__ void __launch_bounds__(256)
gemm_wmma_f32(const float* __restrict__ A, int lda,
              const float* __restrict__ W, int K,
              const float* __restrict__ bias,
              const float* __restrict__ addsrc,
              float* __restrict__ out_raw,
              float* __restrict__ out_act,
              int M, int N)
{
  const int lane = threadIdx.x & 31;
  const int wave = threadIdx.x >> 5;
  const int m0   = blockIdx.x * 256 + wave * 32;
  if (m0 >= M) return;                       // wave-uniform
  const int n0   = blockIdx.y * 64;

  const int ml   = lane & 15;                // A row / B,C column within tile
  const int kh   = lane >> 4;                // k-half select
  const int koff = kh * 2;

  int r0 = m0 + ml;      if (r0 >= M) r0 = M - 1;
  int r1 = m0 + 16 + ml; if (r1 >= M) r1 = M - 1;
  const float* Ap0 = A + (size_t)r0 * lda + koff;
  const float* Ap1 = A + (size_t)r1 * lda + koff;

  int c0 = n0 + 0 * 16 + ml; if (c0 >= N) c0 = N - 1;
  int c1 = n0 + 1 * 16 + ml; if (c1 >= N) c1 = N - 1;
  int c2 = n0 + 2 * 16 + ml; if (c2 >= N) c2 = N - 1;
  int c3 = n0 + 3 * 16 + ml; if (c3 >= N) c3 = N - 1;
  const float* Wp0 = W + (size_t)c0 * K + koff;
  const float* Wp1 = W + (size_t)c1 * K + koff;
  const float* Wp2 = W + (size_t)c2 * K + koff;
  const float* Wp3 = W + (size_t)c3 * K + koff;

  v8f acc00 = {}, acc01 = {}, acc02 = {}, acc03 = {};
  v8f acc10 = {}, acc11 = {}, acc12 = {}, acc13 = {};

  // prologue: fragments for k=0
  v2f a0 = *(const v2f*)(Ap0);
  v2f a1 = *(const v2f*)(Ap1);
  v2f b0 = *(const v2f*)(Wp0);
  v2f b1 = *(const v2f*)(Wp1);
  v2f b2 = *(const v2f*)(Wp2);
  v2f b3 = *(const v2f*)(Wp3);

  for (int k0 = 0; k0 < K - 4; k0 += 4) {
    // prefetch next k-step while current WMMAs run
    const int kn = k0 + 4;
    v2f na0 = *(const v2f*)(Ap0 + kn);
    v2f na1 = *(const v2f*)(Ap1 + kn);
    v2f nb0 = *(const v2f*)(Wp0 + kn);
    v2f nb1 = *(const v2f*)(Wp1 + kn);
    v2f nb2 = *(const v2f*)(Wp2 + kn);
    v2f nb3 = *(const v2f*)(Wp3 + kn);

    acc00 = wmma_f32(a0, b0, acc00);
    acc01 = wmma_f32(a0, b1, acc01);
    acc02 = wmma_f32(a0, b2, acc02);
    acc03 = wmma_f32(a0, b3, acc03);
    acc10 = wmma_f32(a1, b0, acc10);
    acc11 = wmma_f32(a1, b1, acc11);
    acc12 = wmma_f32(a1, b2, acc12);
    acc13 = wmma_f32(a1, b3, acc13);

    a0 = na0; a1 = na1; b0 = nb0; b1 = nb1; b2 = nb2; b3 = nb3;
  }
  // epilogue: last k-step
  acc00 = wmma_f32(a0, b0, acc00);
  acc01 = wmma_f32(a0, b1, acc01);
  acc02 = wmma_f32(a0, b2, acc02);
  acc03 = wmma_f32(a0, b3, acc03);
  acc10 = wmma_f32(a1, b0, acc10);
  acc11 = wmma_f32(a1, b1, acc11);
  acc12 = wmma_f32(a1, b2, acc12);
  acc13 = wmma_f32(a1, b3, acc13);

  // C/D layout: VGPR v, lanes 0-15 -> row v, lanes 16-31 -> row v+8; col = lane&15
  v8f accs[2][4] = {{acc00, acc01, acc02, acc03}, {acc10, acc11, acc12, acc13}};
#pragma unroll
  for (int s = 0; s < 2; ++s) {
#pragma unroll
    for (int t = 0; t < 4; ++t) {
      const int col = n0 + t * 16 + ml;
      if (col >= N) continue;
      const float bi = bias ? bias[col] : 0.f;
#pragma unroll
      for (int v = 0; v < 8; ++v) {
        const int row = m0 + s * 16 + kh * 8 + v;
        if (row >= M) continue;
        float val = accs[s][t][v] + bi;
        if (addsrc) val += addsrc[(size_t)row * N + col];
        if (out_raw) out_raw[(size_t)row * N + col] = val;
        if (out_act) out_act[(size_t)row * N + col] = (val > 0.f) ? val : 0.f;
      }
    }
  }
}

// ---------------------------------------------------------------------
// out[row, col_off + c] = sum_{j<15} msg[graph[row,j], c]  for c<HIDDEN,
// and 0 for pad columns. One wave per row.
// ---------------------------------------------------------------------
__global__ void __launch_bounds__(256)
gather_sum_kernel(const float* __restrict__ msg,
                  const int* __restrict__ graph,
                  float* __restrict__ out, int rows,
                  int stride, int col_off)
{
  const int wave = threadIdx.x >> 5;
  const int lane = threadIdx.x & 31;
  const int row  = blockIdx.x * 8 + wave;
  if (row >= rows) return;
  const int* g = graph + (size_t)row * MAX_NB;
  int idx[MAX_NB];
#pragma unroll
  for (int j = 0; j < MAX_NB; ++j) idx[j] = g[j];
  const int width = stride - col_off;              // HIDDEN..HIDDEN+3
  float* orow = out + (size_t)row * stride + col_off;
  for (int c = lane; c < width; c += 32) {
    float acc = 0.f;
    if (c < HIDDEN) {
#pragma unroll
      for (int j = 0; j < MAX_NB; ++j)
        acc += msg[(size_t)idx[j] * HIDDEN + c];
    }
    orow[c] = acc;                                 // pad columns get exact 0
  }
}

// Wp[n,k<K] = W[n,k]; Wp[n,K<=k<Kp] = 0
__global__ void __launch_bounds__(256)
pad_weights(const float* __restrict__ W, float* __restrict__ Wp, int N, int K, int Kp)
{
  const int i = blockIdx.x * 256 + threadIdx.x;
  if (i >= N * Kp) return;
  const int n = i / Kp, k = i - n * Kp;
  Wp[i] = (k < K) ? W[(size_t)n * K + k] : 0.f;
}

// ainput[r, 0..34] = fatoms[r, :]   (stride KO_PAD)
__global__ void __launch_bounds__(256)
pack_fatoms(const float* __restrict__ f, float* __restrict__ ainput, int M)
{
  const int i = blockIdx.x * 256 + threadIdx.x;
  if (i >= M * ATOM_FDIM) return;
  const int r = i / ATOM_FDIM, c = i - r * ATOM_FDIM;
  ainput[(size_t)r * KO_PAD + c] = f[i];
}

// ---------------------------------------------------------------------
// deterministic column stats (two-stage), then normalize+relu+pool
// ---------------------------------------------------------------------
__global__ void __launch_bounds__(256)
colstats_partial(const float* __restrict__ h,
                 float* __restrict__ psum, float* __restrict__ psq, int Mrows)
{
  const int col   = blockIdx.x * 256 + threadIdx.x;   // < 512
  const int chunk = blockIdx.y;
  const int per   = (Mrows + NCHUNKS - 1) / NCHUNKS;
  const int r0 = chunk * per;
  int r1 = r0 + per; if (r1 > Mrows) r1 = Mrows;
  float s = 0.f, s2 = 0.f;
  if (col < HIDDEN) {
    for (int r = r0; r < r1; ++r) {
      float v = h[(size_t)r * HIDDEN + col];
      s += v; s2 += v * v;
    }
  }
  psum[chunk * NCOLPAD + col] = s;
  psq [chunk * NCOLPAD + col] = s2;
}

__global__ void __launch_bounds__(256)
colstats_reduce(const float* __restrict__ psum, const float* __restrict__ psq,
                float* __restrict__ meanvar, int Mrows)
{
  const int col = blockIdx.x * 256 + threadIdx.x;
  if (col >= HIDDEN) return;
  float s = 0.f, s2 = 0.f;
  for (int ch = 0; ch < NCHUNKS; ++ch) {
    s  += psum[ch * NCOLPAD + col];
    s2 += psq [ch * NCOLPAD + col];
  }
  const float mean = s / (float)Mrows;
  const float var  = s2 / (float)Mrows - mean * mean;
  meanvar[col]           = mean;
  meanvar[NCOLPAD + col] = var;
}

__global__ void __launch_bounds__(256)
norm_pool_kernel(const float* __restrict__ h,
                 const float* __restrict__ meanvar,
                 const float* __restrict__ gamma,
                 const float* __restrict__ beta,
                 const long long* __restrict__ scope,
                 float* __restrict__ out, int Mrows)
{
  const int mol = blockIdx.x;
  const long long start = scope[(size_t)mol * 2 + 0];
  const long long len   = scope[(size_t)mol * 2 + 1];
  for (int c = threadIdx.x; c < HIDDEN; c += blockDim.x) {
    const float mean = meanvar[c];
    const float var  = meanvar[NCOLPAD + c];
    const float rstd = rsqrtf(var + 1e-5f);
    const float g = gamma[c], b = beta[c];
    float acc = 0.f;
    for (long long r = start; r < start + len; ++r) {
      if (r < 0 || r >= Mrows) continue;
      float v = (h[(size_t)r * HIDDEN + c] - mean) * rstd * g + b;
      acc += (v > 0.f) ? v : 0.f;
    }
    out[(size_t)mol * HIDDEN + c] = (len > 0) ? (acc / (float)len) : 0.f;
  }
}

// ---------------------------------------------------------------------
extern "C" void kernel_launch(void* const* d_in, const int* in_sizes, int n_in,
                              void* d_out, int out_size, void* d_ws, size_t ws_size,
                              hipStream_t stream) {
  (void)in_sizes; (void)n_in; (void)out_size; (void)ws_size;
  const float*     fatoms   = (const float*)d_in[0];
  const float*     fbonds   = (const float*)d_in[1];
  const int*       agraph   = (const int*)d_in[2];
  const int*       bgraph   = (const int*)d_in[3];
  const long long* scope    = (const long long*)d_in[4];
  const float*     tree_msg = (const float*)d_in[5];
  const float*     W_i      = (const float*)d_in[6];
  const float*     W_h      = (const float*)d_in[7];
  const float*     W_o      = (const float*)d_in[8];
  const float*     b_o      = (const float*)d_in[9];
  const float*     gamma    = (const float*)d_in[10];
  const float*     beta     = (const float*)d_in[11];
  float*           out      = (float*)d_out;

  // -------- workspace layout (floats, all region sizes even) --------
  float* ws = (float*)d_ws;
  const size_t msg_sz    = (size_t)N_MSGROWS * HIDDEN;   // 31,500,450
  const size_t binput_sz = (size_t)N_BONDS   * HIDDEN;   // 27,000,000
  const size_t nei_sz    = (size_t)N_BONDS   * KH_PAD;   // 27,120,000
  float* msg      = ws;                                  // [70001,450]
  float* msg_gm   = msg + (size_t)N_MESS * HIDDEN;       // graph_message rows
  float* binput   = msg + (msg_sz + 2);                  // [60000,450] (even offset)
  float* nei      = binput + binput_sz;                  // [60000,452] padded
  float* ainput   = nei;                                 // [30000,488] (reuses dead nei)
  float* h        = binput;                              // [30000,450] (reuses dead binput)
  float* W_hp     = nei + nei_sz;                        // [450,452]
  float* W_op     = W_hp + (size_t)HIDDEN * KH_PAD;      // [450,488]
  float* psum     = W_op + (size_t)HIDDEN * KO_PAD;      // [32,512]
  float* psq      = psum + (size_t)NCHUNKS * NCOLPAD;    // [32,512]
  float* meanvar  = psq  + (size_t)NCHUNKS * NCOLPAD;    // [2,512]

  dim3 blk(256, 1, 1);

  // message[0:N_MESS] = tree_message ; pad weights for tail-free K loops
  hipMemcpyAsync(msg, tree_msg, (size_t)N_MESS * HIDDEN * sizeof(float),
                 hipMemcpyDeviceToDevice, stream);
  pad_weights<<<dim3((HIDDEN * KH_PAD + 255) / 256), blk, 0, stream>>>(
      W_h, W_hp, HIDDEN, HIDDEN, KH_PAD);
  pad_weights<<<dim3((HIDDEN * KO_PAD + 255) / 256), blk, 0, stream>>>(
      W_o, W_op, HIDDEN, ATOM_FDIM + HIDDEN, KO_PAD);

  // binput = fbonds @ W_i^T ; graph_message = relu(binput)   (K=40, no pad needed)
  {
    dim3 grid((N_BONDS + 255) / 256, (HIDDEN + 63) / 64, 1);
    gemm_wmma_f32<<<grid, blk, 0, stream>>>(fbonds, BONDF, W_i, BONDF,
                                            nullptr, nullptr,
                                            binput, msg_gm, N_BONDS, HIDDEN);
  }

  // DEPTH-1 message passing iterations
  for (int it = 0; it < 3; ++it) {
    gather_sum_kernel<<<dim3((N_BONDS + 7) / 8), blk, 0, stream>>>(
        msg, bgraph, nei, N_BONDS, KH_PAD, 0);
    dim3 grid((N_BONDS + 255) / 256, (HIDDEN + 63) / 64, 1);
    gemm_wmma_f32<<<grid, blk, 0, stream>>>(nei, KH_PAD, W_hp, KH_PAD,
                                            nullptr, binput,
                                            nullptr, msg_gm, N_BONDS, HIDDEN);
  }

  // ainput = [fatoms | gather(agraph)]  (packed, stride 488, zero pad cols)
  pack_fatoms<<<dim3((N_ATOMS * ATOM_FDIM + 255) / 256), blk, 0, stream>>>(
      fatoms, ainput, N_ATOMS);
  gather_sum_kernel<<<dim3((N_ATOMS + 7) / 8), blk, 0, stream>>>(
      msg, agraph, ainput, N_ATOMS, KO_PAD, ATOM_FDIM);

  // h = ainput @ W_o^T + b_o
  {
    dim3 grid((N_ATOMS + 255) / 256, (HIDDEN + 63) / 64, 1);
    gemm_wmma_f32<<<grid, blk, 0, stream>>>(ainput, KO_PAD, W_op, KO_PAD,
                                            b_o, nullptr,
                                            h, nullptr, N_ATOMS, HIDDEN);
  }

  // batch-norm stats (deterministic 2-stage)
  colstats_partial<<<dim3(2, NCHUNKS), blk, 0, stream>>>(h, psum, psq, N_ATOMS);
  colstats_reduce <<<dim3(2), blk, 0, stream>>>(psum, psq, meanvar, N_ATOMS);

  // normalize + relu + per-molecule mean pooling
  norm_pool_kernel<<<dim3(N_MOLS), blk, 0, stream>>>(
      h, meanvar, gamma, beta, scope, out, N_ATOMS);
}